// HGTLayer_6511170421667
// MI455X (gfx1250) — compile-verified
//
#include <hip/hip_runtime.h>
#include <math.h>

// ---------------- problem constants ----------------
#define DH     256       // model dim
#define NHEAD  4
#define DKH    64
#define N_PED  40000
#define N_LANE 8000
#define E_ADJ  600000
#define E_ON   300000

typedef float v2f __attribute__((ext_vector_type(2)));
typedef float v8f __attribute__((ext_vector_type(8)));

// ---------------- helpers ----------------
__device__ inline unsigned enc_f(float f) {
    unsigned u = __float_as_uint(f);
    return (u & 0x80000000u) ? ~u : (u | 0x80000000u);
}
__device__ inline float dec_f(unsigned u) {
    return (u & 0x80000000u) ? __uint_as_float(u & 0x7fffffffu)
                             : __uint_as_float(~u);
}

// ---------------------------------------------------------------
// Fuse relation transform into projection weight:
//   F[i][h*64+c] = sum_d W[h*64+d][i] * rel[h][d][c]   (F is K-major: Out = X @ F)
//   fb[h*64+c]   = sum_d b[h*64+d]   * rel[h][d][c]
// ---------------------------------------------------------------
__global__ void fuse_weight_kernel(const float* __restrict__ W,
                                   const float* __restrict__ bvec,
                                   const float* __restrict__ rel,
                                   float* __restrict__ F,
                                   float* __restrict__ fb) {
    const int i = blockIdx.x;        // input dim (k of the GEMM)
    const int j = threadIdx.x;       // output dim
    const int h = j >> 6;
    const int c = j & 63;
    const float* relc = rel + h * DKH * DKH + c;   // stride 64 over d
    float s = 0.f;
    #pragma unroll 4
    for (int d = 0; d < DKH; ++d)
        s += W[(h * DKH + d) * DH + i] * relc[d * DKH];
    F[i * DH + j] = s;
    if (i == 0) {
        float sb = 0.f;
        #pragma unroll 4
        for (int d = 0; d < DKH; ++d)
            sb += bvec[h * DKH + d] * relc[d * DKH];
        fb[j] = sb;
    }
}

// B[k][j] = W[j][k]  (turn torch Linear weight into K-major operand)
__global__ void transpose256(const float* __restrict__ W, float* __restrict__ B) {
    const int k = blockIdx.x;
    const int j = threadIdx.x;
    B[k * DH + j] = W[j * DH + k];
}

// ---------------------------------------------------------------
// Out[N][256] = a_scale * X[N][256] @ B[256][256] + bias[256]
// optional epilogue: Out = alpha*Out + (1-alpha)*hres, alpha = sigmoid(skip[skip_idx])
//
// One wave computes a 32x256 output strip (two 16-row M tiles sharing every
// B operand) via V_WMMA_F32_16X16X4_F32.  B is staged through LDS in K-chunks
// of 64 with a K-pair interleaved layout so each B operand is a single
// aligned 8-byte LDS read straight into an even VGPR pair (no shuffles).
//   Bsh[(p*256 + col)*2 + {0,1}] = B[kc+2p+{0,1}][col],  p = K-pair index
// ---------------------------------------------------------------
#define KC 64
__global__ __launch_bounds__(256)
void gemm_wmma(const float* __restrict__ X, const float* __restrict__ B,
               const float* __restrict__ bias, float* __restrict__ Out,
               int Mtiles32, float a_scale,
               const float* __restrict__ hres, const float* __restrict__ skip,
               int skip_idx) {
    __shared__ float Bsh[KC * DH];               // 64 KB
    const int lane = threadIdx.x & 31;
    const int wave = threadIdx.x >> 5;
    const int mt   = blockIdx.x * 8 + wave;      // 32-row tile index
    const int r    = lane & 15;                  // row-in-tile / col-in-tile
    const int half = lane >> 4;                  // K-pair select
    const bool active = (mt < Mtiles32);
    const int row_base = mt * 32;

    v8f acc0[16], acc1[16];
    #pragma unroll
    for (int j = 0; j < 16; ++j)
        #pragma unroll
        for (int q = 0; q < 8; ++q) { acc0[j][q] = 0.f; acc1[j][q] = 0.f; }

    for (int kc = 0; kc < DH; kc += KC) {
        __syncthreads();
        // cooperative staging: pair-interleave rows kc+2p, kc+2p+1 (p=0..31)
        for (int u = threadIdx.x; u < 32 * 128; u += 256) {
            const int p = u >> 7;                 // K-pair 0..31
            const int c = (u & 127) * 2;          // column, step 2
            const float* g0 = B + (size_t)(kc + 2 * p) * DH + c;
            const float2 t0 = *(const float2*)g0;          // row kc+2p
            const float2 t1 = *(const float2*)(g0 + DH);   // row kc+2p+1
            float4 w;
            w.x = t0.x; w.y = t1.x; w.z = t0.y; w.w = t1.y;
            *(float4*)(Bsh + ((size_t)p * DH + c) * 2) = w;
        }
        __syncthreads();
        if (active) {
            for (int kk = 0; kk < KC; kk += 4) {
                // A tile 16x4 fp32: lanes 0-15 -> K0/K1, lanes 16-31 -> K2/K3
                const int kb = kc + kk + 2 * half;         // even
                const float* xp0 = X + (size_t)(row_base + r) * DH + kb;
                v2f a0 = *(const v2f*)xp0;                 // rows 0..15 of strip
                v2f a1 = *(const v2f*)(xp0 + 16 * DH);     // rows 16..31
                a0 = a0 * a_scale;
                a1 = a1 * a_scale;
                const int p = (kk + 2 * half) >> 1;        // K-pair in chunk
                const float* bbase = Bsh + ((size_t)p * DH + r) * 2;
                #pragma unroll
                for (int j = 0; j < 16; ++j) {
                    const v2f b = *(const v2f*)(bbase + j * 32);  // {K,K+1} at col
                    acc0[j] = __builtin_amdgcn_wmma_f32_16x16x4_f32(
                        false, a0, false, b, (short)0, acc0[j], false, false);
                    acc1[j] = __builtin_amdgcn_wmma_f32_16x16x4_f32(
                        false, a1, false, b, (short)0, acc1[j], false, false);
                }
            }
        }
    }
    if (!active) return;

    float alpha = 1.f, beta = 0.f;
    if (skip) {
        float sv = skip[skip_idx];
        alpha = 1.f / (1.f + __expf(-sv));
        beta  = 1.f - alpha;
    }
    #pragma unroll
    for (int m = 0; m < 2; ++m) {
        #pragma unroll
        for (int j = 0; j < 16; ++j) {
            #pragma unroll
            for (int i = 0; i < 8; ++i) {
                const int row = row_base + m * 16 + i + 8 * half;  // C/D layout
                const int col = j * 16 + r;
                float v = (m ? acc1[j][i] : acc0[j][i]) + bias[col];
                if (hres) v = alpha * v + beta * hres[(size_t)row * DH + col];
                Out[(size_t)row * DH + col] = v;
            }
        }
    }
}

// ---------------------------------------------------------------
// per-(edge,head) score = dot64(q[dst], k[src]) * rel_pri[h] / sqrt(64)
// + atomic segment-max into smax (order-preserving uint encoding)
// ---------------------------------------------------------------
__global__ void edge_score_kernel(const float* __restrict__ q,
                                  const float* __restrict__ k,
                                  const int* __restrict__ src,
                                  const int* __restrict__ dst, int E,
                                  const float* __restrict__ pri,
                                  float* __restrict__ score,
                                  unsigned* __restrict__ smax) {
    const int tid = blockIdx.x * blockDim.x + threadIdx.x;
    if (tid >= E * NHEAD) return;
    const int e = tid >> 2, h = tid & 3;
    const int si = src[e], di = dst[e];
    const float4* qv = (const float4*)(q + (size_t)di * DH + h * DKH);
    const float4* kv = (const float4*)(k + (size_t)si * DH + h * DKH);
    float acc = 0.f;
    #pragma unroll
    for (int t = 0; t < 16; ++t) {
        float4 a = qv[t], b = kv[t];
        acc += a.x * b.x + a.y * b.y + a.z * b.z + a.w * b.w;
    }
    acc *= pri[h] * 0.125f;                 // 1/sqrt(DK) = 0.125
    score[tid] = acc;
    __hip_atomic_fetch_max(&smax[(size_t)di * NHEAD + h], enc_f(acc),
                           __ATOMIC_RELAXED, __HIP_MEMORY_SCOPE_AGENT);
}

// ex = exp(score - smax[dst]); denom[dst] += ex  (score buffer overwritten with ex)
__global__ void edge_exp_kernel(const int* __restrict__ dst, int E,
                                float* __restrict__ score,
                                const unsigned* __restrict__ smax,
                                float* __restrict__ denom) {
    const int tid = blockIdx.x * blockDim.x + threadIdx.x;
    if (tid >= E * NHEAD) return;
    const int e = tid >> 2, h = tid & 3;
    const int di = dst[e];
    const float m = dec_f(smax[(size_t)di * NHEAD + h]);
    const float ex = __expf(score[tid] - m);
    score[tid] = ex;
    __hip_atomic_fetch_add(&denom[(size_t)di * NHEAD + h], ex,
                           __ATOMIC_RELAXED, __HIP_MEMORY_SCOPE_AGENT);
}

// t[dst] += v[src] * (ex / denom[dst])   (one thread per (edge, float4-chunk))
__global__ void edge_msg_kernel(const float* __restrict__ v,
                                const int* __restrict__ src,
                                const int* __restrict__ dst, int E,
                                const float* __restrict__ ex,
                                const float* __restrict__ denom,
                                float* __restrict__ t) {
    const long tid = (long)blockIdx.x * blockDim.x + threadIdx.x;
    if (tid >= (long)E * 64) return;
    const int e = (int)(tid >> 6);
    const int c4 = (int)(tid & 63);
    const int h = c4 >> 4;
    const int si = src[e], di = dst[e];
    const float coef = ex[(size_t)e * NHEAD + h] / denom[(size_t)di * NHEAD + h];
    const float4 vv = ((const float4*)(v + (size_t)si * DH))[c4];
    float* tp = t + (size_t)di * DH + c4 * 4;
    __hip_atomic_fetch_add(tp + 0, vv.x * coef, __ATOMIC_RELAXED, __HIP_MEMORY_SCOPE_AGENT);
    __hip_atomic_fetch_add(tp + 1, vv.y * coef, __ATOMIC_RELAXED, __HIP_MEMORY_SCOPE_AGENT);
    __hip_atomic_fetch_add(tp + 2, vv.z * coef, __ATOMIC_RELAXED, __HIP_MEMORY_SCOPE_AGENT);
    __hip_atomic_fetch_add(tp + 3, vv.w * coef, __ATOMIC_RELAXED, __HIP_MEMORY_SCOPE_AGENT);
}

// ---------------------------------------------------------------
extern "C" void kernel_launch(void* const* d_in, const int* in_sizes, int n_in,
                              void* d_out, int out_size, void* d_ws, size_t ws_size,
                              hipStream_t stream) {
    const float* h_ped   = (const float*)d_in[0];
    const float* h_lane  = (const float*)d_in[1];
    const float* Wk      = (const float*)d_in[2];
    const float* bk      = (const float*)d_in[3];
    const float* Wq      = (const float*)d_in[4];
    const float* bq      = (const float*)d_in[5];
    const float* Wv      = (const float*)d_in[6];
    const float* bv      = (const float*)d_in[7];
    const float* Wa      = (const float*)d_in[8];
    const float* ba      = (const float*)d_in[9];
    const float* rel_pri = (const float*)d_in[10];
    const float* rel_att = (const float*)d_in[11];
    const float* rel_msg = (const float*)d_in[12];
    const float* skip    = (const float*)d_in[13];
    const int* adj_src   = (const int*)d_in[14];
    const int* adj_dst   = (const int*)d_in[15];
    const int* on_src    = (const int*)d_in[16];
    const int* on_dst    = (const int*)d_in[17];
    const int* ron_src   = (const int*)d_in[18];
    const int* ron_dst   = (const int*)d_in[19];
    float* out = (float*)d_out;

    // -------- workspace layout (floats) --------
    float* ws = (float*)d_ws;
    size_t o = 0;
    float* bufK  = ws + o; o += (size_t)N_PED * DH;
    float* bufV  = ws + o; o += (size_t)N_PED * DH;
    float* bufQp = ws + o; o += (size_t)N_PED * DH;
    float* bufQl = ws + o; o += (size_t)N_LANE * DH;
    float* t_ped = ws + o; o += (size_t)N_PED * DH;
    float* t_lane= ws + o; o += (size_t)N_LANE * DH;
    float* FK    = ws + o; o += DH * DH;
    float* FV    = ws + o; o += DH * DH;
    float* fbK   = ws + o; o += DH;
    float* fbV   = ws + o; o += DH;
    float* BqP   = ws + o; o += DH * DH;
    float* BqL   = ws + o; o += DH * DH;
    float* BaP   = ws + o; o += DH * DH;
    float* BaL   = ws + o; o += DH * DH;
    float* score = ws + o; o += (size_t)E_ADJ * NHEAD;  // reused for all edge types
    unsigned* smax = (unsigned*)(ws + o); o += (size_t)N_PED * NHEAD;
    float* denom = ws + o; o += (size_t)N_PED * NHEAD;
    (void)ws_size; (void)n_in; (void)in_sizes; (void)out_size;

    const int MT_PED = N_PED / 32;    // 1250 (32-row strips)
    const int MT_LANE = N_LANE / 32;  // 250
    const int GB_PED = (MT_PED + 7) / 8;   // 157 blocks of 8 waves
    const int GB_LANE = (MT_LANE + 7) / 8; // 32
    auto cdiv = [](long a, long b) { return (int)((a + b - 1) / b); };
    const int RELSZ = NHEAD * DKH * DKH;   // 16384
    const int WSZ = DH * DH;               // 65536

    // -------- weight prep --------
    transpose256<<<DH, DH, 0, stream>>>(Wq,       BqP);
    transpose256<<<DH, DH, 0, stream>>>(Wq + WSZ, BqL);
    transpose256<<<DH, DH, 0, stream>>>(Wa,       BaP);
    transpose256<<<DH, DH, 0, stream>>>(Wa + WSZ, BaL);

    hipMemsetAsync(t_ped, 0, (size_t)N_PED * DH * 4, stream);
    hipMemsetAsync(t_lane, 0, (size_t)N_LANE * DH * 4, stream);

    // -------- Q projections --------
    gemm_wmma<<<GB_PED, 256, 0, stream>>>(h_ped, BqP, bq, bufQp, MT_PED, 1.f, nullptr, nullptr, 0);
    gemm_wmma<<<GB_LANE, 256, 0, stream>>>(h_lane, BqL, bq + DH, bufQl, MT_LANE, 1.f, nullptr, nullptr, 0);

    // ===== phase e=0: ped->ped over adj (si=0, di=0), into t_ped =====
    fuse_weight_kernel<<<DH, DH, 0, stream>>>(Wk, bk, rel_att, FK, fbK);
    fuse_weight_kernel<<<DH, DH, 0, stream>>>(Wv, bv, rel_msg, FV, fbV);
    gemm_wmma<<<GB_PED, 256, 0, stream>>>(h_ped, FK, fbK, bufK, MT_PED, 1.f, nullptr, nullptr, 0);
    gemm_wmma<<<GB_PED, 256, 0, stream>>>(h_ped, FV, fbV, bufV, MT_PED, 1.f, nullptr, nullptr, 0);
    hipMemsetAsync(smax, 0, (size_t)N_PED * NHEAD * 4, stream);
    hipMemsetAsync(denom, 0, (size_t)N_PED * NHEAD * 4, stream);
    edge_score_kernel<<<cdiv((long)E_ADJ * NHEAD, 256), 256, 0, stream>>>(
        bufQp, bufK, adj_src, adj_dst, E_ADJ, rel_pri + 0 * NHEAD, score, smax);
    edge_exp_kernel<<<cdiv((long)E_ADJ * NHEAD, 256), 256, 0, stream>>>(
        adj_dst, E_ADJ, score, smax, denom);
    edge_msg_kernel<<<cdiv((long)E_ADJ * 64, 256), 256, 0, stream>>>(
        bufV, adj_src, adj_dst, E_ADJ, score, denom, t_ped);

    // ===== phase e=2: lane->ped over ron (si=1, di=0), into t_ped =====
    fuse_weight_kernel<<<DH, DH, 0, stream>>>(Wk + WSZ, bk + DH, rel_att + 2 * RELSZ, FK, fbK);
    fuse_weight_kernel<<<DH, DH, 0, stream>>>(Wv + WSZ, bv + DH, rel_msg + 2 * RELSZ, FV, fbV);
    gemm_wmma<<<GB_LANE, 256, 0, stream>>>(h_lane, FK, fbK, bufK, MT_LANE, 1.f, nullptr, nullptr, 0);
    gemm_wmma<<<GB_LANE, 256, 0, stream>>>(h_lane, FV, fbV, bufV, MT_LANE, 1.f, nullptr, nullptr, 0);
    hipMemsetAsync(smax, 0, (size_t)N_PED * NHEAD * 4, stream);
    hipMemsetAsync(denom, 0, (size_t)N_PED * NHEAD * 4, stream);
    edge_score_kernel<<<cdiv((long)E_ON * NHEAD, 256), 256, 0, stream>>>(
        bufQp, bufK, ron_src, ron_dst, E_ON, rel_pri + 2 * NHEAD, score, smax);
    edge_exp_kernel<<<cdiv((long)E_ON * NHEAD, 256), 256, 0, stream>>>(
        ron_dst, E_ON, score, smax, denom);
    edge_msg_kernel<<<cdiv((long)E_ON * 64, 256), 256, 0, stream>>>(
        bufV, ron_src, ron_dst, E_ON, score, denom, t_ped);

    // ===== phase e=1: ped->lane over on (si=0, di=1), into t_lane =====
    fuse_weight_kernel<<<DH, DH, 0, stream>>>(Wk, bk, rel_att + 1 * RELSZ, FK, fbK);
    fuse_weight_kernel<<<DH, DH, 0, stream>>>(Wv, bv, rel_msg + 1 * RELSZ, FV, fbV);
    gemm_wmma<<<GB_PED, 256, 0, stream>>>(h_ped, FK, fbK, bufK, MT_PED, 1.f, nullptr, nullptr, 0);
    gemm_wmma<<<GB_PED, 256, 0, stream>>>(h_ped, FV, fbV, bufV, MT_PED, 1.f, nullptr, nullptr, 0);
    hipMemsetAsync(smax, 0, (size_t)N_LANE * NHEAD * 4, stream);
    hipMemsetAsync(denom, 0, (size_t)N_LANE * NHEAD * 4, stream);
    edge_score_kernel<<<cdiv((long)E_ON * NHEAD, 256), 256, 0, stream>>>(
        bufQl, bufK, on_src, on_dst, E_ON, rel_pri + 1 * NHEAD, score, smax);
    edge_exp_kernel<<<cdiv((long)E_ON * NHEAD, 256), 256, 0, stream>>>(
        on_dst, E_ON, score, smax, denom);
    edge_msg_kernel<<<cdiv((long)E_ON * 64, 256), 256, 0, stream>>>(
        bufV, on_src, on_dst, E_ON, score, denom, t_lane);

    // ===== final: out = sigmoid(skip)*(scale*t @ Wa.T + ba) + (1-sigmoid)*h =====
    // t_ped carries msg0+msg2; the reference mean (0.5*) is folded into a_scale.
    gemm_wmma<<<GB_PED, 256, 0, stream>>>(t_ped, BaP, ba, out, MT_PED, 0.5f, h_ped, skip, 0);
    gemm_wmma<<<GB_LANE, 256, 0, stream>>>(t_lane, BaL, ba + DH, out + (size_t)N_PED * DH,
                                           MT_LANE, 1.f, h_lane, skip, 1);
}